// AttentionBasedKNN_18983755448437
// MI455X (gfx1250) — compile-verified
//
#include <hip/hip_runtime.h>
#include <math.h>

typedef float v2f __attribute__((ext_vector_type(2)));
typedef float v8f __attribute__((ext_vector_type(8)));

#define D_FEAT 61
#define DQ     64
#define V_DIM  64
#define K_TOP  16
#define WAVES  4

// ---------------------------------------------------------------------------
// Kernel 1: q[row] = [ x/||x|| (61) , pos/||pos|| (3) ]   (||q||^2 == 2 exactly)
// ---------------------------------------------------------------------------
__global__ void build_q_kernel(const float* __restrict__ x,
                               const float* __restrict__ pos,
                               float* __restrict__ q, int N) {
    int row = blockIdx.x * blockDim.x + threadIdx.x;
    if (row >= N) return;
    float s = 0.f;
    for (int j = 0; j < D_FEAT; ++j) { float v = x[row * D_FEAT + j]; s += v * v; }
    float inv = 1.0f / sqrtf(s);
    for (int j = 0; j < D_FEAT; ++j) q[row * DQ + j] = x[row * D_FEAT + j] * inv;
    float p0 = pos[row * 3 + 0], p1 = pos[row * 3 + 1], p2 = pos[row * 3 + 2];
    float inv2 = 1.0f / sqrtf(p0 * p0 + p1 * p1 + p2 * p2);
    q[row * DQ + 61] = p0 * inv2;
    q[row * DQ + 62] = p1 * inv2;
    q[row * DQ + 63] = p2 * inv2;
}

// ---------------------------------------------------------------------------
// Kernel 2: dbn[j] = ||database[j]||^2
// ---------------------------------------------------------------------------
__global__ void db_norms_kernel(const float* __restrict__ db,
                                float* __restrict__ dbn, int M) {
    int row = blockIdx.x * blockDim.x + threadIdx.x;
    if (row >= M) return;
    float s = 0.f;
    for (int j = 0; j < DQ; ++j) { float v = db[row * DQ + j]; s += v * v; }
    dbn[row] = s;
}

// ---------------------------------------------------------------------------
// Kernel 3: per wave: 16 queries vs all M db rows via v_wmma_f32_16x16x4_f32,
// streaming top-16 selection, softmax, weighted value gather.
// ---------------------------------------------------------------------------
__global__ __launch_bounds__(32 * WAVES)
void knn_attn_main(const float* __restrict__ q,
                   const float* __restrict__ db,
                   const float* __restrict__ dbn,
                   const float* __restrict__ wvals,
                   float* __restrict__ out, int M) {
    __shared__ float s_d2[WAVES][16][17];   // padded to dodge bank conflicts
    __shared__ float s_attn[WAVES][16][K_TOP];
    __shared__ int   s_idx[WAVES][16][K_TOP];

    const int lane  = threadIdx.x & 31;
    const int wv    = threadIdx.x >> 5;
    const int qbase = (blockIdx.x * WAVES + wv) * 16;

    const int mrow = lane & 15;   // row (A) / column (B,C) owned by this lane
    const int ksel = lane >> 4;   // K-half select per ISA 32-bit A/B layout

    // Preload the 16 A fragments (query tile reused across all db tiles).
    // Layout (16x4 f32 A): lanes 0-15 hold K={k0,k0+1}, lanes 16-31 K={k0+2,k0+3}.
    v2f afrag[16];
#pragma unroll
    for (int kk = 0; kk < 16; ++kk)
        afrag[kk] = *reinterpret_cast<const v2f*>(
            &q[(size_t)(qbase + mrow) * DQ + kk * 4 + ksel * 2]);

    float topd[K_TOP];
    int   topi[K_TOP];
#pragma unroll
    for (int i = 0; i < K_TOP; ++i) { topd[i] = 3.4e38f; topi[i] = 0; }
    float worst = 3.4e38f;

    const int ntiles = M >> 4;
    for (int jt = 0; jt < ntiles; ++jt) {
        const int jbase = jt << 4;

        // B fragments: column n = db row (jbase+n); same K split as A.
        v2f bfrag[16];
#pragma unroll
        for (int kk = 0; kk < 16; ++kk)
            bfrag[kk] = *reinterpret_cast<const v2f*>(
                &db[(size_t)(jbase + mrow) * DQ + kk * 4 + ksel * 2]);

        if (jt + 1 < ntiles)
            __builtin_prefetch(&db[(size_t)(jbase + 16 + mrow) * DQ], 0, 0);

        v8f acc = {};
#pragma unroll
        for (int kk = 0; kk < 16; ++kk)
            acc = __builtin_amdgcn_wmma_f32_16x16x4_f32(
                false, afrag[kk], false, bfrag[kk], (short)0, acc, false, false);

        // d2 = ||q||^2 + ||db||^2 - 2 q.db = 2 + dbn - 2*acc
        // C layout: VGPR r -> row (r + 8*ksel), col = lane&15.
        const float dn = dbn[jbase + mrow];
#pragma unroll
        for (int r = 0; r < 8; ++r)
            s_d2[wv][r + 8 * ksel][mrow] = 2.0f + dn - 2.0f * acc[r];

        asm volatile("s_wait_dscnt 0" ::: "memory");   // wave-local LDS RAW

        // Threshold-guarded insertion top-16 (lanes 0-15, one query row each).
        if (lane < 16) {
#pragma unroll 1
            for (int c = 0; c < 16; ++c) {
                float d = s_d2[wv][lane][c];
                if (d < worst) {
                    int p = K_TOP - 1;
                    while (p > 0 && topd[p - 1] > d) {
                        topd[p] = topd[p - 1]; topi[p] = topi[p - 1]; --p;
                    }
                    topd[p] = d; topi[p] = jbase + c;
                    worst = topd[K_TOP - 1];
                }
            }
        }
        asm volatile("s_wait_dscnt 0" ::: "memory");   // before next tile's stores
    }

    // Softmax over scaled = 8 - 4*d2 (sqrt_dk = 8). topd ascending -> max at [0].
    if (lane < 16) {
        float mx = 8.0f - 4.0f * topd[0];
        float e[K_TOP], ssum = 0.f;
#pragma unroll
        for (int i = 0; i < K_TOP; ++i) {
            e[i] = __expf((8.0f - 4.0f * topd[i]) - mx);
            ssum += e[i];
        }
        float inv = 1.0f / ssum;
#pragma unroll
        for (int i = 0; i < K_TOP; ++i) {
            s_attn[wv][lane][i] = e[i] * inv;
            s_idx[wv][lane][i]  = topi[i];
        }
    }
    asm volatile("s_wait_dscnt 0" ::: "memory");

    // Weighted gather: each lane produces 2 of the 64 output dims per query row.
    const int d0 = lane * 2;
    for (int mq = 0; mq < 16; ++mq) {
        float a0 = 0.f, a1 = 0.f;
#pragma unroll
        for (int k = 0; k < K_TOP; ++k) {
            float a  = s_attn[wv][mq][k];
            int   id = s_idx[wv][mq][k];
            v2f w = *reinterpret_cast<const v2f*>(&wvals[(size_t)id * V_DIM + d0]);
            a0 += a * w.x;
            a1 += a * w.y;
        }
        v2f o; o.x = a0; o.y = a1;
        *reinterpret_cast<v2f*>(&out[(size_t)(qbase + mq) * V_DIM + d0]) = o;
    }
}

// ---------------------------------------------------------------------------
extern "C" void kernel_launch(void* const* d_in, const int* in_sizes, int n_in,
                              void* d_out, int out_size, void* d_ws, size_t ws_size,
                              hipStream_t stream) {
    const float* x     = (const float*)d_in[0];
    const float* pos   = (const float*)d_in[1];
    const float* db    = (const float*)d_in[2];
    const float* wvals = (const float*)d_in[3];
    // d_in[4] is topk==16 (hardcoded as K_TOP)

    const int N = in_sizes[0] / D_FEAT;   // 8192
    const int M = in_sizes[2] / DQ;       // 16384

    float* q   = (float*)d_ws;            // N*64 floats  (2 MB)
    float* dbn = q + (size_t)N * DQ;      // M floats     (64 KB)
    float* out = (float*)d_out;

    build_q_kernel<<<(N + 255) / 256, 256, 0, stream>>>(x, pos, q, N);
    db_norms_kernel<<<(M + 255) / 256, 256, 0, stream>>>(db, dbn, M);

    const int qtiles_per_block = WAVES;                       // 64 queries/block
    const int blocks = N / (16 * qtiles_per_block);           // 8192/64 = 128
    knn_attn_main<<<blocks, 32 * WAVES, 0, stream>>>(q, db, dbn, wvals, out, M);
}